// VectorQuantizedLatent_87900800680037
// MI455X (gfx1250) — compile-verified
//
#include <hip/hip_runtime.h>

typedef __attribute__((ext_vector_type(2))) float v2f;
typedef __attribute__((ext_vector_type(4))) float v4f;
typedef __attribute__((ext_vector_type(8))) float v8f;

#define L_TOTAL 32768
#define K_EMB   1024
#define E_DIM   64
#define NTILES  (K_EMB / 16)   // 64 column tiles
#define LDS_STRIDE 68          // padded row stride (17 x float4): conflict-free
                               // ds reads: banks 4p+4j+{0..3} permute 0..63

// ---------------------------------------------------------------------------
// Kernel A: bias[k] = 0.5 * ||e_k||^2   (1024 values; tiny, L2-resident)
// ---------------------------------------------------------------------------
__global__ __launch_bounds__(256) void vq_bias_kernel(const float* __restrict__ emb,
                                                      float* __restrict__ bias) {
  int k = blockIdx.x * 256 + threadIdx.x;
  if (k < K_EMB) {
    const float* row = emb + (size_t)k * E_DIM;
    float s = 0.0f;
#pragma unroll
    for (int j = 0; j < E_DIM; ++j) {
      float v = row[j];
      s += v * v;
    }
    bias[k] = 0.5f * s;
  }
}

// ---------------------------------------------------------------------------
// Kernel B: per-wave 16-row tile; WMMA f32 16x16x4 GEMM vs all 1024 codes.
// Double-buffered, bank-conflict-free LDS staging; global prefetch pinned
// ahead of the WMMA block; B fragments software-pipelined in registers;
// full bias table resident in LDS; per-lane running argmax; gather epilogue.
// Block = 256 threads = 8 waves -> 128 rows per block, 256 blocks.
// ---------------------------------------------------------------------------
__global__ __launch_bounds__(256) void vq_main_kernel(const float* __restrict__ x,
                                                      const float* __restrict__ emb,
                                                      const float* __restrict__ bias,
                                                      float* __restrict__ out) {
  __shared__ float ldsB[2][16 * LDS_STRIDE];   // two padded 16x64 tiles (~8.5 KB)
  __shared__ float ldsBiasAll[K_EMB];          // full bias table (4 KB)

  const int tid  = threadIdx.x;
  const int lane = tid & 31;
  const int wave = tid >> 5;
  const int p    = lane & 15;   // row (A) / column (B,C) position in tile
  const int h    = lane >> 4;   // half-wave selector (K pair / M half)

  const int m0 = (blockIdx.x * 8 + wave) * 16;   // this wave's first latent row

  // Staging map: tid's float4 of the contiguous 1024-float tile -> padded row
  const int srow = tid >> 4;            // embedding row within tile (0..15)
  const int scol = (tid & 15) * 4;      // column within row
  const int sdst = srow * LDS_STRIDE + scol;

  // --- Load A fragments: 16 K-slices of the 16x64 row block ---------------
  // ISA layout (32-bit A 16x4): lanes 0-15 -> M=0..15; VGPR0 = K {0|2},
  // VGPR1 = K {1|3} per half-wave  =>  per lane: consecutive pair at 4j+2h.
  v2f a[16];
  const float* xrow = x + (size_t)(m0 + p) * E_DIM + 2 * h;
#pragma unroll
  for (int j = 0; j < 16; ++j)
    a[j] = *(const v2f*)(xrow + 4 * j);

  float best[8];
  int   bidx[8];
#pragma unroll
  for (int i = 0; i < 8; ++i) { best[i] = -3.402823466e38f; bidx[i] = 0; }

  // --- One-time staging: full bias table + tile 0 ---------------------------
  *(v4f*)(&ldsBiasAll[tid * 4]) = *(const v4f*)(bias + tid * 4);
  *(v4f*)(&ldsB[0][sdst])       = *(const v4f*)(emb + tid * 4);

  int cur = 0;
  for (int nt = 0; nt < NTILES; ++nt) {
    __syncthreads();   // tile `cur` stores visible; prior readers of cur^1 done

    // Issue next tile's global load now; sched_barrier pins it ahead of the
    // WMMA block so its L2 latency is hidden behind the matrix work.
    const int ntn = (nt + 1 < NTILES) ? nt + 1 : NTILES - 1;   // uniform clamp
    v4f nstage = *(const v4f*)(emb + (size_t)ntn * (16 * E_DIM) + tid * 4);
    __builtin_amdgcn_sched_barrier(0);

    // --- 16x16 tile GEMM: two accumulator chains, B pipelined one pair deep
    const float* brow = ldsB[cur] + p * LDS_STRIDE + 2 * h;  // B[k][n]=emb[n][k]
    v2f b0 = *(const v2f*)(brow + 0);
    v2f b1 = *(const v2f*)(brow + 4);
    v8f c0 = {};
    v8f c1 = {};
#pragma unroll
    for (int j = 0; j < 8; ++j) {
      const v2f cb0 = b0;
      const v2f cb1 = b1;
      const int jn = (j + 1) & 7;            // wraps on last iter (dead load)
      b0 = *(const v2f*)(brow + 8 * jn);
      b1 = *(const v2f*)(brow + 8 * jn + 4);
      c0 = __builtin_amdgcn_wmma_f32_16x16x4_f32(
          false, a[2 * j],     false, cb0, (short)0, c0, false, false);
      c1 = __builtin_amdgcn_wmma_f32_16x16x4_f32(
          false, a[2 * j + 1], false, cb1, (short)0, c1, false, false);
    }

    // score = x·e - 0.5||e||^2 ; running argmax (strict > keeps first index)
    const int   kcol = nt * 16 + p;
    const float bv   = ldsBiasAll[kcol];   // same addr across half-waves: bcast
#pragma unroll
    for (int i = 0; i < 8; ++i) {
      float s = (c0[i] + c1[i]) - bv;
      if (s > best[i]) { best[i] = s; bidx[i] = kcol; }
    }

    // Store prefetched tile into the alternate buffer (dead on last iter).
    *(v4f*)(&ldsB[cur ^ 1][sdst]) = nstage;
    cur ^= 1;
  }

  // --- Cross-lane argmax within each 16-lane group (xor masks stay inside) -
#pragma unroll
  for (int i = 0; i < 8; ++i) {
#pragma unroll
    for (int off = 8; off >= 1; off >>= 1) {
      float ov = __shfl_xor(best[i], off, 32);
      int   oi = __shfl_xor(bidx[i], off, 32);
      if (ov > best[i] || (ov == best[i] && oi < bidx[i])) {
        best[i] = ov;
        bidx[i] = oi;
      }
    }
  }
  // Lanes 0-15 hold argmax for rows m0+0..7 (reg i -> row m0+i),
  // lanes 16-31 for rows m0+8..15.

  float* outX = out;
  float* outQ = out + (size_t)L_TOTAL * E_DIM;
  float* outZ = out + 2 * (size_t)L_TOTAL * E_DIM;
  float* outI = out + 3 * (size_t)L_TOTAL * E_DIM;

  if (p == 0) {
#pragma unroll
    for (int i = 0; i < 8; ++i)
      outI[m0 + h * 8 + i] = (float)bidx[i];
  }

  // --- Gather winning embeddings, write x / q / z_hat (2 floats per lane) --
#pragma unroll
  for (int i = 0; i < 16; ++i) {
    const int src = (i >> 3) << 4;                 // lane 0 or lane 16
    const int kk  = __shfl(bidx[i & 7], src, 32);  // broadcast row i's index
    const size_t r = (size_t)(m0 + i) * E_DIM;

    v2f q  = *(const v2f*)(emb + (size_t)kk * E_DIM + lane * 2);
    v2f xv = *(const v2f*)(x + r + lane * 2);
    v2f z;
    z.x = (xv.x + q.x) - xv.x;   // exact STE arithmetic order as reference
    z.y = (xv.y + q.y) - xv.y;

    *(v2f*)(outX + r + lane * 2) = xv;
    *(v2f*)(outQ + r + lane * 2) = q;
    *(v2f*)(outZ + r + lane * 2) = z;
  }
}

// ---------------------------------------------------------------------------
extern "C" void kernel_launch(void* const* d_in, const int* in_sizes, int n_in,
                              void* d_out, int out_size, void* d_ws, size_t ws_size,
                              hipStream_t stream) {
  (void)in_sizes; (void)n_in; (void)out_size; (void)ws_size;

  const float* x   = (const float*)d_in[0];   // [L_TOTAL * E_DIM]
  const float* emb = (const float*)d_in[1];   // [K_EMB, E_DIM]
  float* out  = (float*)d_out;
  float* bias = (float*)d_ws;                 // 1024 floats of scratch

  vq_bias_kernel<<<dim3(K_EMB / 256), dim3(256), 0, stream>>>(emb, bias);
  vq_main_kernel<<<dim3(L_TOTAL / 128), dim3(256), 0, stream>>>(x, emb, bias, out);
}